// QConv_32246614458474
// MI455X (gfx1250) — compile-verified
//
#include <hip/hip_runtime.h>
#include <hip/hip_bf16.h>

// ---------------------------------------------------------------------------
// QConv: per-pixel 32-vector (27 patch vals + eps, zero padded), y = W v with
// W = [Re U(0:16,:); Im U(0:16,:)] (U = fixed 5-qubit circuit unitary),
// out[c] = 8*(y[c]^2 + y[c+16]^2)/||v||^2.  GEMM mapped to WMMA f32 16x16x4.
// ---------------------------------------------------------------------------

typedef __attribute__((ext_vector_type(2))) float v2f;
typedef __attribute__((ext_vector_type(8))) float v8f;

#define HW      50176      // 224*224
#define IMW     224
#define NPIX    401408     // 8*224*224  (divisible by 128; 50176 = 392*128)
#define EPSV    0.001f
#define VSTRIDE 34         // padded row stride of V tile in LDS (bank-conflict free)
#define WSTRIDE 34         // padded row stride of W in LDS
#define OSTRIDE 132        // padded row stride of out tile (float4-aligned, low conflict)

// --- Kernel 1: build the 32x32 real matrix W from the 15 rotation angles ----
// Thread k applies the circuit to basis state |k>, yielding column U[:,k].
__global__ void build_W_kernel(const float* __restrict__ wts, float* __restrict__ Wout) {
    const int k = threadIdx.x;
    if (k >= 32) return;
    float re[32], im[32];
#pragma unroll
    for (int i = 0; i < 32; ++i) { re[i] = 0.f; im[i] = 0.f; }
    re[k] = 1.f;

    // Rot(phi,theta,omega) on qubit q (qubit 0 = MSB, bit 4-q):
    //   U00=(cap - i sap)c  U01=-(cam + i sam)s
    //   U10=(cam - i sam)s  U11=(cap + i sap)c
#pragma unroll
    for (int q = 0; q < 5; ++q) {
        const float phi = wts[q * 3 + 0];
        const float th  = wts[q * 3 + 1];
        const float om  = wts[q * 3 + 2];
        const float c   = __cosf(0.5f * th),        s   = __sinf(0.5f * th);
        const float cap = __cosf(0.5f * (phi + om)), sap = __sinf(0.5f * (phi + om));
        const float cam = __cosf(0.5f * (phi - om)), sam = __sinf(0.5f * (phi - om));
        const float u00r =  cap * c, u00i = -sap * c;
        const float u01r = -cam * s, u01i = -sam * s;
        const float u10r =  cam * s, u10i = -sam * s;
        const float u11r =  cap * c, u11i =  sap * c;
        const int bit = 1 << (4 - q);
#pragma unroll
        for (int i = 0; i < 32; ++i) {
            if (i & bit) continue;
            const int j = i | bit;
            const float a0r = re[i], a0i = im[i], a1r = re[j], a1i = im[j];
            re[i] = u00r * a0r - u00i * a0i + u01r * a1r - u01i * a1i;
            im[i] = u00r * a0i + u00i * a0r + u01r * a1i + u01i * a1r;
            re[j] = u10r * a0r - u10i * a0i + u11r * a1r - u11i * a1i;
            im[j] = u10r * a0i + u10i * a0r + u11r * a1i + u11i * a1r;
        }
    }
    // CNOT ring: control q -> target (q+1)%5; swap amplitudes i <-> i^tbit where cbit set
#pragma unroll
    for (int q = 0; q < 5; ++q) {
        const int cb = 1 << (4 - q);
        const int tb = 1 << (4 - ((q + 1) % 5));
#pragma unroll
        for (int i = 0; i < 32; ++i) {
            if (!(i & cb)) continue;
            if (i & tb) continue;
            const int j = i ^ tb;
            float tr = re[i], ti = im[i];
            re[i] = re[j]; im[i] = im[j];
            re[j] = tr;    im[j] = ti;
        }
    }
#pragma unroll
    for (int c = 0; c < 16; ++c) {
        Wout[c * 32 + k]        = re[c];  // rows 0..15  : Re U[0:16, k]
        Wout[(c + 16) * 32 + k] = im[c];  // rows 16..31 : Im U[0:16, k]
    }
}

// --- Kernel 2: main. 128 pixels / block, 8 waves, WMMA f32 16x16x4 ----------
__global__ __launch_bounds__(256) void qconv_main_kernel(const float* __restrict__ x,
                                                         const float* __restrict__ Wm,
                                                         float* __restrict__ out) {
    __shared__ float Vt[128 * VSTRIDE];   // V tile: [pixel][feature]
    __shared__ float Wl[32 * WSTRIDE];    // W:      [row n][feature k]
    __shared__ float scl[128];            // 8 / ||v||^2 per pixel (precomputed)
    __shared__ float Ot[16 * OSTRIDE];    // out tile: [channel][pixel]

    const int t     = threadIdx.x;
    const int pbase = blockIdx.x * 128;   // blocks align exactly with batches
    const int b     = pbase / HW;
    const int s0    = pbase - b * HW;     // linear h*224+w offset within batch

    // stage W into LDS (padded rows)
    for (int i = t; i < 1024; i += 256)
        Wl[(i >> 5) * WSTRIDE + (i & 31)] = Wm[i];

    // Phase 1: stencil gather -> V tile (feature f = c*9 + 3*dy + dx, +eps; f>=27 -> 0)
#pragma unroll
    for (int rep = 0; rep < 16; ++rep) {
        const int e = rep * 256 + t;          // 0..4095
        const int m = e >> 5;                 // local pixel
        const int f = e & 31;                 // feature
        float val = 0.f;
        if (f < 27) {
            const int c  = f / 9;
            const int rr = (f - c * 9) / 3;
            const int cc = f - c * 9 - rr * 3;
            const int s  = s0 + m;
            const int h  = s / IMW;
            const int w  = s - h * IMW;
            const int hh = h + rr - 1, ww = w + cc - 1;
            float xv = 0.f;
            if ((unsigned)hh < 224u && (unsigned)ww < 224u)
                xv = x[(((size_t)b * 3 + c) * 224 + hh) * 224 + ww];
            val = xv + EPSV;
        }
        Vt[m * VSTRIDE + f] = val;
    }
    __syncthreads();

    // norms -> store 8/||v||^2 once per pixel (128 divides/block instead of
    // a per-lane IEEE divide sequence in the epilogue)
    if (t < 128) {
        float acc = 0.f;
#pragma unroll
        for (int f = 0; f < 27; ++f) { const float v = Vt[t * VSTRIDE + f]; acc += v * v; }
        scl[t] = 8.0f / acc;
    }
    __syncthreads();

    // Phase 2: per-wave 16x32 output tile via 16 chained WMMA f32 16x16x4.
    // A (16x4 f32): lanes 0-15 hold M=0..15 K={4kk,4kk+1}; lanes 16-31 K={4kk+2,4kk+3}
    // B (4x16 f32): lanes 0-15 hold N=0..15 K={4kk,4kk+1}; lanes 16-31 K={4kk+2,4kk+3}
    // C/D (16x16):  VGPR r, lanes 0-15 -> M=r, lanes 16-31 -> M=r+8
    const int wid  = t >> 5;
    const int lane = t & 31;
    const int lo   = lane & 15;
    const int hi   = lane >> 4;

    v8f acc0 = {};   // y_real = Re(Uv)[0..15]
    v8f acc1 = {};   // y_imag = Im(Uv)[0..15]
#pragma unroll
    for (int kk = 0; kk < 8; ++kk) {
        const int k0 = kk * 4 + 2 * hi;
        const v2f a  = *(const v2f*)&Vt[(wid * 16 + lo) * VSTRIDE + k0];
        const v2f b0 = *(const v2f*)&Wl[lo * WSTRIDE + k0];
        const v2f b1 = *(const v2f*)&Wl[(16 + lo) * WSTRIDE + k0];
        acc0 = __builtin_amdgcn_wmma_f32_16x16x4_f32(false, a, false, b0, (short)0, acc0, false, false);
        acc1 = __builtin_amdgcn_wmma_f32_16x16x4_f32(false, a, false, b1, (short)0, acc1, false, false);
    }

    // out[c] = (yr^2+yi^2) * (8/||v||^2) ; write transposed tile [channel][pixel]
#pragma unroll
    for (int r = 0; r < 8; ++r) {
        const float yr = acc0[r], yi = acc1[r];
        const int   m  = wid * 16 + r + 8 * hi;     // local pixel
        Ot[lo * OSTRIDE + m] = (yr * yr + yi * yi) * scl[m];   // channel = lo
    }
    __syncthreads();

    // Phase 3: coalesced float4 stores, contiguous per channel
    float* gbase = out + (size_t)b * (16 * (size_t)HW) + s0;
    for (int i = t; i < 512; i += 256) {              // 16 ch * 32 float4
        const int c = i >> 5;
        const int q = i & 31;
        const float4 v4 = *(const float4*)&Ot[c * OSTRIDE + q * 4];
        *(float4*)(gbase + (size_t)c * HW + q * 4) = v4;
    }
}

extern "C" void kernel_launch(void* const* d_in, const int* in_sizes, int n_in,
                              void* d_out, int out_size, void* d_ws, size_t ws_size,
                              hipStream_t stream) {
    const float* x   = (const float*)d_in[0];   // (8,3,224,224)
    const float* wts = (const float*)d_in[1];   // (1,5,3)
    float* out = (float*)d_out;                 // (8,16,224,224)
    float* Wm  = (float*)d_ws;                  // 32x32 f32 = 4KB scratch

    build_W_kernel<<<1, 32, 0, stream>>>(wts, Wm);
    qconv_main_kernel<<<NPIX / 128, 256, 0, stream>>>(x, Wm, out);
}